// RelGraphEmbedLayer_7009386627525
// MI455X (gfx1250) — compile-verified
//
#include <hip/hip_runtime.h>
#include <hip/hip_bf16.h>

// CDNA5 / gfx1250, wave32.
// out[i,:] = (tids[i]==0) ? feat0[type_ids[i],:] @ W0 : table[node_ids[i],:]
//
// Projection uses V_WMMA_F32_16X16X4_F32 (full fp32 precision, matches the
// fp32 reference). One wave computes a 16x128 output tile: 8 v8f accumulators,
// 64 K-chunks of 4, 8 WMMAs per chunk.

typedef __attribute__((ext_vector_type(2))) float v2f;
typedef __attribute__((ext_vector_type(8))) float v8f;

#define D_IN   256
#define EMBED  128
#define PHASE_CHUNKS 32           // chunks staged per LDS phase (K=128)
#define LDS_V2F (PHASE_CHUNKS * 2 * EMBED)  // 8192 v2f = 64 KB

// ---------------------------------------------------------------------------
// Embedding path: tid != 0 rows are a pure gather+copy. 8 rows per block,
// 32 lanes per row, float4 per lane -> fully coalesced 512B row copies.
// ---------------------------------------------------------------------------
__global__ __launch_bounds__(256) void embed_copy_kernel(
    const int* __restrict__ node_ids, const int* __restrict__ node_tids,
    const float* __restrict__ table, float* __restrict__ out, int N) {
  const int row = blockIdx.x * 8 + (threadIdx.x >> 5);
  const int c = (threadIdx.x & 31) * 4;
  if (row < N && node_tids[row] != 0) {
    const size_t nid = (size_t)node_ids[row];
    const float4 v = *(const float4*)(table + nid * EMBED + c);
    *(float4*)(out + (size_t)row * EMBED + c) = v;
  }
}

// ---------------------------------------------------------------------------
// Projection path: gathered-A GEMM on the fp32 WMMA pipe.
//
// Per block: 8 waves, 128 rows. W0 (256x128 fp32 = 128 KB) is staged through
// 64 KB of LDS in two K-phases, pre-swizzled into B-fragment layout so the
// inner loop's B fetch is one ds_load_b64 per fragment (the compiler pairs
// them into ds_load_2addr_b64):
//   sB[(c*2 + h)*128 + n] = { W0[(4c+2h)*128 + n], W0[(4c+2h+1)*128 + n] }
// matching the 16x16x4 fp32 B layout (lanes 0-15: K=4c,4c+1;
// lanes 16-31: K=4c+2,4c+3 -- same half-split as the A/CD layouts).
//
// A fragment: unconditional per-lane float2 from the gathered feat0 row.
// Rows with tid!=0 are redirected to feat0 row 0: their D rows are garbage
// but never stored (row m of D depends only on row m of A), and row 0 is a
// single shared 1 KB line that stays L2-resident -- no HBM waste, and no
// per-chunk exec-mask branching around the load.
// ---------------------------------------------------------------------------
__global__ __launch_bounds__(256) void proj_wmma_kernel(
    const int* __restrict__ node_tids, const int* __restrict__ type_ids,
    const float* __restrict__ feat0, const float* __restrict__ W0,
    float* __restrict__ out, int N) {
  __shared__ v2f sB[LDS_V2F];

  const int t    = threadIdx.x;
  const int w    = t >> 5;        // wave in block (0..7)
  const int lane = t & 31;
  const int hf   = lane >> 4;     // half-wave (selects K sub-pair)
  const int r    = lane & 15;     // row within 16-row tile / col within 16-col tile
  const int base = blockIdx.x * 128 + w * 16;

  int gi = base + r;
  if (gi >= N) gi = N - 1;        // clamp: WMMA needs all lanes live; stores are guarded
  const int my_tid = node_tids[gi];
  // tid!=0 rows -> feat0 row 0 (L2-resident dummy); their outputs aren't stored.
  const size_t typ = (my_tid == 0) ? (size_t)type_ids[gi] : 0;
  const float* arow = feat0 + typ * D_IN + 2 * hf;

  v8f acc[8] = {};                // 16x128 fp32 tile, 64 VGPRs

  for (int phase = 0; phase < 2; ++phase) {
    // ---- stage + swizzle this phase's half of W0 into LDS (coalesced in n)
    const float* wsrc = W0 + (size_t)phase * (PHASE_CHUNKS * 4) * EMBED;
    for (int e = t; e < LDS_V2F; e += 256) {
      const int c = e >> 8;             // chunk within phase
      const int h = (e >> 7) & 1;       // half
      const int n = e & (EMBED - 1);    // output column
      const int krow = c * 4 + h * 2;
      v2f v;
      v.x = wsrc[krow * EMBED + n];
      v.y = wsrc[(krow + 1) * EMBED + n];
      sB[e] = v;
    }
    __syncthreads();

    // ---- K loop over this phase: 32 chunks x 8 column tiles of WMMA
    const float* ap = arow + phase * (PHASE_CHUNKS * 4);
#pragma unroll 4
    for (int cl = 0; cl < PHASE_CHUNKS; ++cl) {
      const v2f a = *(const v2f*)(ap + cl * 4);      // global_load_b64
      const v2f* bp = &sB[(cl * 2 + hf) * EMBED + r];
#pragma unroll
      for (int j = 0; j < 8; ++j) {
        const v2f b = bp[j * 16];                    // ds_load_b64 (paired 2addr)
        acc[j] = __builtin_amdgcn_wmma_f32_16x16x4_f32(
            /*neg_a=*/false, a, /*neg_b=*/false, b,
            /*c_mod=*/(short)0, acc[j], /*reuse_a=*/false, /*reuse_b=*/false);
      }
    }
    __syncthreads();                                 // protect LDS before restage
  }

  // ---- writeback: C/D layout -> VGPR p holds rows {p, p+8}, cols = lane%16.
  // Only tid==0 rows are ours; embed_copy_kernel owns the rest (disjoint).
#pragma unroll
  for (int p = 0; p < 8; ++p) {
    const int row = base + p + 8 * hf;
    if (row < N && node_tids[row] == 0) {
      float* orow = out + (size_t)row * EMBED + r;
#pragma unroll
      for (int j = 0; j < 8; ++j) orow[j * 16] = acc[j][p];
    }
  }
}

// ---------------------------------------------------------------------------
extern "C" void kernel_launch(void* const* d_in, const int* in_sizes, int n_in,
                              void* d_out, int out_size, void* d_ws, size_t ws_size,
                              hipStream_t stream) {
  const int*   node_ids   = (const int*)d_in[0];
  const int*   node_tids  = (const int*)d_in[1];
  const int*   type_ids   = (const int*)d_in[2];
  const float* feat0      = (const float*)d_in[3];
  const float* W0         = (const float*)d_in[4];
  const float* table      = (const float*)d_in[5];
  float*       out        = (float*)d_out;
  const int N = in_sizes[0];

  // Disjoint row ownership: order between the two kernels doesn't matter.
  embed_copy_kernel<<<(N + 7) / 8, 256, 0, stream>>>(node_ids, node_tids,
                                                     table, out, N);
  proj_wmma_kernel<<<(N + 127) / 128, 256, 0, stream>>>(node_tids, type_ids,
                                                        feat0, W0, out, N);
}